// EnhancedSAGELayer_48928267436146
// MI455X (gfx1250) — compile-verified
//
#include <hip/hip_runtime.h>

#define NNODES 50000
#define NEDGES 512000
#define DIM    128
#define NTYPE  3
#define WPB    2      // waves per block in the GEMM kernel

typedef __attribute__((ext_vector_type(2))) float v2f;
typedef __attribute__((ext_vector_type(8))) float v8f;

// ---------------------------------------------------------------------------
// Kernel 1: zero the aggregation workspace (agg[T][N][D] + cnt[T][N])
// ---------------------------------------------------------------------------
__global__ void sage_zero_kernel(float* __restrict__ p, size_t n) {
  size_t i = (size_t)blockIdx.x * blockDim.x + threadIdx.x;
  size_t stride = (size_t)gridDim.x * blockDim.x;
  for (; i < n; i += stride) p[i] = 0.0f;
}

// ---------------------------------------------------------------------------
// Kernel 2: one wave (32 lanes) per (type, edge). Gather x[src] as float4 and
// scatter-add into agg[t][dst]. x (25.6MB) and agg (76.8MB) are L2-resident
// on MI455X (192MB L2), so this is an L2-bandwidth / atomic-rate problem.
// ---------------------------------------------------------------------------
__global__ void sage_scatter_kernel(const float* __restrict__ x,
                                    const int* __restrict__ ei0,
                                    const int* __restrict__ ei1,
                                    const int* __restrict__ ei2,
                                    float* __restrict__ agg,
                                    float* __restrict__ cnt) {
  int gw = blockIdx.x * (blockDim.x >> 5) + (threadIdx.x >> 5);
  int lane = threadIdx.x & 31;
  if (gw >= NTYPE * NEDGES) return;
  int t = gw / NEDGES;
  int e = gw - t * NEDGES;
  const int* ei = (t == 0) ? ei0 : (t == 1) ? ei1 : ei2;
  int src = ei[e];            // edge_index[0][e]
  int dst = ei[NEDGES + e];   // edge_index[1][e]

  const float4* xr = (const float4*)(x + (size_t)src * DIM);
  float4 v = xr[lane];  // 32 lanes * 4 floats = 128 features

  float* ar = agg + ((size_t)t * NNODES + dst) * DIM + lane * 4;
  atomicAdd(ar + 0, v.x);
  atomicAdd(ar + 1, v.y);
  atomicAdd(ar + 2, v.z);
  atomicAdd(ar + 3, v.w);
  if (lane == 0) atomicAdd(cnt + (size_t)t * NNODES + dst, 1.0f);
}

// ---------------------------------------------------------------------------
// Kernel 3: fused per-16-node-tile WMMA pipeline, phase-split to keep
// register pressure low (acc[8] and the output accumulator never coexist).
//
// Phase 1 (per type t): h_t = mean_t @ Wl_t^T + x @ Wr_t^T + bl_t,
//   row-normalize * att[t], stage scaled h_t into LDS (row-major [16][128]).
// Phase 2: out = sum_t h_t @ Wc_t^T + bc, one v8f accumulator per stripe.
//
// f32 WMMA fragment layouts (ISA 7.12.2):
//   A 16x4 : lane L -> row L%16, K = 4*kc + 2*(L/16) + {0,1} in v2f
//   B 4x16 : lane L -> col L%16, same K pairing (mirror of A)
//   C/D    : lane L, vgpr v -> row v + 8*(L/16), col L%16
// ---------------------------------------------------------------------------
__global__ void sage_gemm_kernel(const float* __restrict__ x,
                                 const float* __restrict__ agg,
                                 const float* __restrict__ cnt,
                                 const float* __restrict__ Wl,
                                 const float* __restrict__ bl,
                                 const float* __restrict__ Wr,
                                 const float* __restrict__ att,
                                 const float* __restrict__ Wc,
                                 const float* __restrict__ bc,
                                 float* __restrict__ out) {
  __shared__ float lds_h[WPB][NTYPE][16 * DIM];  // 2 waves * 24KB = 48KB

  const int wave = threadIdx.x >> 5;
  const int lane = threadIdx.x & 31;
  const int tile = blockIdx.x * WPB + wave;
  if (tile >= NNODES / 16) return;  // wave-uniform exit; EXEC stays all-ones

  const int r0 = tile * 16;
  const int ln = lane & 15;   // A-row / B-col / D-col for this lane
  const int hh = lane >> 4;   // half select: K offset 2*hh, D-row offset 8*hh
  const int arow = r0 + ln;

  const float* xrow = x + (size_t)arow * DIM;

  // ---------------- Phase 1: build the three scaled h_t tiles ----------------
  for (int t = 0; t < NTYPE; ++t) {
    float c0 = cnt[(size_t)t * NNODES + arow];
    float invc = 1.0f / fmaxf(c0, 1.0f);
    const float* aggrow = agg + ((size_t)t * NNODES + arow) * DIM;
    const float* WlT = Wl + (size_t)t * DIM * DIM;
    const float* WrT = Wr + (size_t)t * DIM * DIM;
    const float attv = att[t];
    float* hb = &lds_h[wave][t][0];

    v8f acc[8];
#pragma unroll
    for (int n = 0; n < 8; ++n) acc[n] = (v8f)0.0f;

#pragma unroll 1
    for (int kc = 0; kc < 32; ++kc) {
      const int k = kc * 4 + hh * 2;
      // A fragments loaded once, reused across all 8 output stripes
      float2 am = *(const float2*)(aggrow + k);
      float2 ax = *(const float2*)(xrow + k);
      v2f a1, a2;
      a1.x = am.x * invc; a1.y = am.y * invc;
      a2.x = ax.x;        a2.y = ax.y;
#pragma unroll
      for (int ncol = 0; ncol < 8; ++ncol) {
        const int nb = ncol * 16 + ln;
        float2 bw1 = *(const float2*)(WlT + (size_t)nb * DIM + k);
        float2 bw2 = *(const float2*)(WrT + (size_t)nb * DIM + k);
        v2f b1, b2;
        b1.x = bw1.x; b1.y = bw1.y;
        b2.x = bw2.x; b2.y = bw2.y;
        acc[ncol] = __builtin_amdgcn_wmma_f32_16x16x4_f32(
            false, a1, false, b1, (short)0, acc[ncol], false, false);
        acc[ncol] = __builtin_amdgcn_wmma_f32_16x16x4_f32(
            false, a2, false, b2, (short)0, acc[ncol], false, false);
      }
    }

    // bias + row sum-of-squares partials
    float s[8];
#pragma unroll
    for (int v = 0; v < 8; ++v) s[v] = 0.0f;
#pragma unroll
    for (int ncol = 0; ncol < 8; ++ncol) {
      float bb = bl[t * DIM + ncol * 16 + ln];
#pragma unroll
      for (int v = 0; v < 8; ++v) {
        float hv = acc[ncol][v] + bb;
        acc[ncol][v] = hv;
        s[v] += hv * hv;
      }
    }
    // reduce across the 16 lanes of each half (rows v+8*hh)
#pragma unroll
    for (int m = 1; m < 16; m <<= 1)
#pragma unroll
      for (int v = 0; v < 8; ++v) s[v] += __shfl_xor(s[v], m, 16);

    float scl[8];
#pragma unroll
    for (int v = 0; v < 8; ++v)
      scl[v] = attv / fmaxf(sqrtf(s[v]), 1e-12f);

    // stage scaled h_t to LDS in row-major [16][128] (C/D -> A relayout);
    // scl[v] is exactly the scale of the row being written.
#pragma unroll
    for (int ncol = 0; ncol < 8; ++ncol)
#pragma unroll
      for (int v = 0; v < 8; ++v)
        hb[(v + 8 * hh) * DIM + ncol * 16 + ln] = acc[ncol][v] * scl[v];
  }

  // ---------------- Phase 2: out = sum_t h_t @ Wc_t^T + bc ----------------
  const float* hwave = &lds_h[wave][0][0];
#pragma unroll 1
  for (int ncol = 0; ncol < 8; ++ncol) {
    const int nb = ncol * 16 + ln;
    v8f c = (v8f)0.0f;
#pragma unroll 1
    for (int t = 0; t < NTYPE; ++t) {
      const float* wc_row = Wc + (size_t)nb * (NTYPE * DIM) + t * DIM;
      const float* ht = hwave + t * (16 * DIM);
#pragma unroll 1
      for (int kc = 0; kc < 32; ++kc) {
        const int k = kc * 4 + hh * 2;
        float2 ah = *(const float2*)(ht + ln * DIM + k);
        float2 bw = *(const float2*)(wc_row + k);
        v2f a, b;
        a.x = ah.x; a.y = ah.y;
        b.x = bw.x; b.y = bw.y;
        c = __builtin_amdgcn_wmma_f32_16x16x4_f32(
            false, a, false, b, (short)0, c, false, false);
      }
    }
    float bb = bc[nb];
#pragma unroll
    for (int v = 0; v < 8; ++v)
      out[(size_t)(r0 + v + 8 * hh) * DIM + nb] = c[v] + bb;
  }
}

// ---------------------------------------------------------------------------
extern "C" void kernel_launch(void* const* d_in, const int* in_sizes, int n_in,
                              void* d_out, int out_size, void* d_ws, size_t ws_size,
                              hipStream_t stream) {
  const float* x   = (const float*)d_in[0];
  const int*   ei0 = (const int*)d_in[1];
  const int*   ei1 = (const int*)d_in[2];
  const int*   ei2 = (const int*)d_in[3];
  const float* Wl  = (const float*)d_in[4];
  const float* bl  = (const float*)d_in[5];
  const float* Wr  = (const float*)d_in[6];
  const float* att = (const float*)d_in[7];
  const float* Wc  = (const float*)d_in[8];
  const float* bc  = (const float*)d_in[9];
  float* out = (float*)d_out;

  float* agg = (float*)d_ws;                              // [T][N][D]
  float* cnt = agg + (size_t)NTYPE * NNODES * DIM;        // [T][N]

  size_t zn = (size_t)NTYPE * NNODES * DIM + (size_t)NTYPE * NNODES;
  sage_zero_kernel<<<2048, 256, 0, stream>>>(agg, zn);

  int nwaves = NTYPE * NEDGES;            // one wave per (type, edge)
  int sblocks = nwaves / 8;               // 256 thr = 8 waves per block
  sage_scatter_kernel<<<sblocks, 256, 0, stream>>>(x, ei0, ei1, ei2, agg, cnt);

  int tiles = NNODES / 16;                // 3125
  int gblocks = (tiles + WPB - 1) / WPB;  // 1563 blocks of 2 waves
  sage_gemm_kernel<<<gblocks, WPB * 32, 0, stream>>>(x, agg, cnt, Wl, bl, Wr,
                                                     att, Wc, bc, out);
}